// HadamardModule_68152541053199
// MI455X (gfx1250) — compile-verified
//
#include <hip/hip_runtime.h>

// y = H128 * x / sqrt(128) per contiguous 128-group, Sylvester ordering.
// H128 = H8 (x) H16 (Kronecker, i = i1*16 + i0).
// Per 16x16 tile (two adjacent groups, 1 KiB): D = X * (s*H16) via four chained
// v_wmma_f32_16x16x4_f32 (full fp32 precision, matching the fp32 reference),
// then FWHT-8 across the 8 accumulator VGPRs (rows 0-7/8-15 live in VGPRs 0-7
// on the two lane halves -> H8 stage is pure per-lane VALU, zero cross-lane).
// Persistent waves loop over uniform (readfirstlane) tile ranges so addressing
// is saddr+voffset with SALU induction, and the B operand is built once.

typedef __attribute__((ext_vector_type(2))) float v2f;
typedef __attribute__((ext_vector_type(8))) float v8f;

__global__ __launch_bounds__(256, 1)
void had128_wmma_kernel(const float* __restrict__ x, float* __restrict__ y,
                        int num_tiles, int tiles_per_wave) {
  const int lane = threadIdx.x & 31;
  const int wave_id = blockIdx.x * 8 + (threadIdx.x >> 5);

  int t0 = wave_id * tiles_per_wave;
  int t_end = t0 + tiles_per_wave;
  if (t_end > num_tiles) t_end = num_tiles;
  // Force uniform loop bounds into SGPRs: scalar loop, EXEC untouched.
  const int t0u = __builtin_amdgcn_readfirstlane(t0);
  const int t1u = __builtin_amdgcn_readfirstlane(t_end);
  if (t0u >= t1u) return;

  const int m = lane & 15;           // A-layout row / B-layout column (N)
  const int kh2 = (lane >> 4) << 1;  // K sub-offset: 0 (lanes 0-15), 2 (16-31)

  // B operand: (s*H16)[4k+Kc][N], sign = (-1)^popcount(K&N), s = 2^-3.5.
  // Built once per wave; sign xor'ed into the fp32 sign bit.
  const int si = __float_as_int(0.08838834764831845f);
  v2f b[4];
#pragma unroll
  for (int k = 0; k < 4; ++k) {
    const int j0 = 4 * k + kh2;
    b[k].x = __int_as_float(si ^ ((__popc(j0 & m) & 1) << 31));
    b[k].y = __int_as_float(si ^ ((__popc((j0 + 1) & m) & 1) << 31));
  }

  // Loop-invariant per-lane element offsets (voffset); tile base is uniform.
  const int lx = m * 16 + kh2;              // load offset within tile
  const int ly = (lane >> 4) * 128 + m;     // store offset within tile

  for (int ti = t0u; ti < t1u; ++ti) {
    const float* __restrict__ bx = x + (size_t)ti * 256;  // uniform -> saddr
    float* __restrict__ by = y + (size_t)ti * 256;        // uniform -> saddr

    v8f acc = {};
#pragma unroll
    for (int k = 0; k < 4; ++k) {
      // A chunk k: X[:, 4k..4k+3] in fp32 A-layout -> one b64 load per lane
      v2f a = *(const v2f*)(bx + lx + 4 * k);
      acc = __builtin_amdgcn_wmma_f32_16x16x4_f32(
          /*neg_a=*/false, a, /*neg_b=*/false, b[k],
          /*c_mod=*/(short)0, acc, /*reuse_a=*/false, /*reuse_b=*/false);
    }

    // FWHT-8 (Sylvester), stages reordered bit2->bit1->bit0 so the first two
    // stages pair adjacent accumulator VGPRs (mov-free v_pk_add_f32).
    v2f a01 = {acc[0], acc[1]}, a23 = {acc[2], acc[3]};
    v2f a45 = {acc[4], acc[5]}, a67 = {acc[6], acc[7]};
    v2f u01 = a01 + a45, u23 = a23 + a67;   // bit2
    v2f u45 = a01 - a45, u67 = a23 - a67;
    v2f w01 = u01 + u23, w23 = u01 - u23;   // bit1
    v2f w45 = u45 + u67, w67 = u45 - u67;
    float r0 = w01.x + w01.y, r1 = w01.x - w01.y;  // bit0 (scalar)
    float r2 = w23.x + w23.y, r3 = w23.x - w23.y;
    float r4 = w45.x + w45.y, r5 = w45.x - w45.y;
    float r6 = w67.x + w67.y, r7 = w67.x - w67.y;

    // D-layout store: row j of group (lane>>4) -> g*128 + j*16 + m.
    // Non-temporal: written once, never re-read.
    float* o = by + ly;
    __builtin_nontemporal_store(r0, o + 0 * 16);
    __builtin_nontemporal_store(r1, o + 1 * 16);
    __builtin_nontemporal_store(r2, o + 2 * 16);
    __builtin_nontemporal_store(r3, o + 3 * 16);
    __builtin_nontemporal_store(r4, o + 4 * 16);
    __builtin_nontemporal_store(r5, o + 5 * 16);
    __builtin_nontemporal_store(r6, o + 6 * 16);
    __builtin_nontemporal_store(r7, o + 7 * 16);
  }
}

extern "C" void kernel_launch(void* const* d_in, const int* in_sizes, int n_in,
                              void* d_out, int out_size, void* d_ws, size_t ws_size,
                              hipStream_t stream) {
  (void)n_in; (void)d_ws; (void)ws_size; (void)out_size;
  const float* x = (const float*)d_in[0];
  float* y = (float*)d_out;

  const long long n = (long long)in_sizes[0];  // 8192*4096, divisible by 256
  const int num_tiles = (int)(n / 256);        // 131072 tiles of 1 KiB

  int blocks = 2048;                            // 16384 persistent waves
  if (blocks * 8 > num_tiles) blocks = (num_tiles + 7) / 8;
  if (blocks < 1) blocks = 1;
  const int waves = blocks * 8;
  const int tiles_per_wave = (num_tiles + waves - 1) / waves;

  had128_wmma_kernel<<<blocks, 256, 0, stream>>>(x, y, num_tiles, tiles_per_wave);
}